// EMDLayer_15118284882651
// MI455X (gfx1250) — compile-verified
//
#include <hip/hip_runtime.h>

// EMD / Sinkhorn layer for MI455X (gfx1250, wave32, WMMA).
// E=4, Q=75, W=5, N=64  ->  B = 1500 independent 64x64 Sinkhorn solves.
// One workgroup (128 threads = 4 waves) per batch; K resident in LDS;
// matvecs done with V_WMMA_F32_16X16X4_F32 (fp32 required: K spans [2e-9, 1]).

#define QD      75
#define WD      5
#define NDIM    64
#define STRIDE  65          // LDS row pad -> conflict-free row & column access
#define ITERS   50
#define INV_EPS 20.0f       // 1 / 0.05
#define TEMP    12.5f
#define WEPS    1e-5f

typedef __attribute__((ext_vector_type(2))) float v2f;
typedef __attribute__((ext_vector_type(8))) float v8f;

__device__ __forceinline__ float wave_max32(float m) {
#pragma unroll
  for (int off = 16; off > 0; off >>= 1)
    m = fmaxf(m, __shfl_xor(m, off, 32));
  return m;
}

__device__ __forceinline__ float wave_sum32(float m) {
#pragma unroll
  for (int off = 16; off > 0; off >>= 1)
    m += __shfl_xor(m, off, 32);
  return m;
}

__global__ __launch_bounds__(128) void emd_sinkhorn_kernel(
    const float* __restrict__ sim,    // [B, 64, 64]
    const float* __restrict__ w1,     // [E,Q,W,64] == [B,64]
    const float* __restrict__ w2,     // [E,W,Q,64]
    float* __restrict__ out)          // [B]
{
  __shared__ float Ksh[NDIM * STRIDE];               // exp((sim-1)/eps)
  __shared__ float la[NDIM], lb[NDIM];               // log of normalized marginals
  __shared__ float lu[NDIM], lv[NDIM];               // Sinkhorn potentials
  __shared__ float vex[NDIM];                        // exp(pot - max)
  __shared__ float red[4];

  const int tid  = threadIdx.x;
  const int lane = tid & 31;
  const int wv   = tid >> 5;
  const int b    = blockIdx.x;

  // batch -> (e,q,w); weight_2 is [E,W,Q,N] transposed to [E,Q,W,N]
  const int e  = b / (QD * WD);
  const int rr = b - e * (QD * WD);
  const int q  = rr / WD;
  const int w  = rr - q * WD;

  const float* simb = sim + (size_t)b * (NDIM * NDIM);
  const float* w1b  = w1  + (size_t)b * NDIM;
  const float* w2b  = w2  + (size_t)((e * WD + w) * QD + q) * NDIM;

  // ---- build K in LDS (stride-65 padded) ----
#pragma unroll
  for (int t = 0; t < (NDIM * NDIM) / 128; ++t) {
    int idx = t * 128 + tid;
    int i = idx >> 6, j = idx & 63;
    Ksh[i * STRIDE + j] = __expf((simb[idx] - 1.0f) * INV_EPS);
  }
  // ---- marginals: relu + eps, renormalize to sum N, take log ----
  if (tid < NDIM) {
    la[tid] = fmaxf(w1b[tid], 0.0f) + WEPS;
    lb[tid] = fmaxf(w2b[tid], 0.0f) + WEPS;
    lv[tid] = 0.0f;
  }
  __syncthreads();
  float sa = wave_sum32(la[lane] + la[lane + 32]);   // each wave redundantly
  float sb = wave_sum32(lb[lane] + lb[lane + 32]);
  __syncthreads();
  if (tid < NDIM) {
    la[tid] = __logf(la[tid]) + __logf(64.0f / sa);
    lb[tid] = __logf(lb[tid]) + __logf(64.0f / sb);
  }
  __syncthreads();

  const int m0   = wv * 16;        // this wave's output block
  const int mrow = lane & 15;
  const int h    = lane >> 4;      // K-half select (ISA A/B 32-bit layout)

  for (int it = 0; it < ITERS; ++it) {
    // ======== row pass: r = K * exp(lv - mv);  lu = la - (log r + mv) ========
    float mv = wave_max32(fmaxf(lv[lane], lv[lane + 32]));
    if (tid < NDIM) vex[tid] = __expf(lv[tid] - mv);
    __syncthreads();

    v8f acc = {};
#pragma unroll
    for (int k0 = 0; k0 < NDIM; k0 += 4) {
      v2f a, bb;
      // A = K block: A[M][k] = Ksh[(m0+M)*65 + k0+k]; VGPR0->k=2h, VGPR1->k=2h+1
      a.x  = Ksh[(m0 + mrow) * STRIDE + k0 + 2 * h];
      a.y  = Ksh[(m0 + mrow) * STRIDE + k0 + 2 * h + 1];
      // B = ve replicated across all 16 columns
      bb.x = vex[k0 + 2 * h];
      bb.y = vex[k0 + 2 * h + 1];
      acc = __builtin_amdgcn_wmma_f32_16x16x4_f32(
          false, a, false, bb, (short)0, acc, false, false);
    }
    // D layout: lane l, VGPR v -> row m0 + v + 8*(l>>4); all columns identical
    if (mrow == 0) {
#pragma unroll
      for (int v = 0; v < 8; ++v) {
        int row = m0 + v + 8 * h;
        lu[row] = la[row] - (__logf(acc[v]) + mv);
      }
    }
    __syncthreads();

    // ======== col pass: c = K^T * exp(lu - mu); lv = lb - (log c + mu) ========
    float mu = wave_max32(fmaxf(lu[lane], lu[lane + 32]));
    if (tid < NDIM) vex[tid] = __expf(lu[tid] - mu);
    __syncthreads();

    v8f acc2 = {};
#pragma unroll
    for (int k0 = 0; k0 < NDIM; k0 += 4) {
      v2f a, bb;
      // A = K^T block: A[M=j][k=i] = Ksh[i*65 + (m0+j)]
      a.x  = Ksh[(k0 + 2 * h) * STRIDE + m0 + mrow];
      a.y  = Ksh[(k0 + 2 * h + 1) * STRIDE + m0 + mrow];
      bb.x = vex[k0 + 2 * h];
      bb.y = vex[k0 + 2 * h + 1];
      acc2 = __builtin_amdgcn_wmma_f32_16x16x4_f32(
          false, a, false, bb, (short)0, acc2, false, false);
    }
    if (mrow == 0) {
#pragma unroll
      for (int v = 0; v < 8; ++v) {
        int col = m0 + v + 8 * h;
        lv[col] = lb[col] - (__logf(acc2[v]) + mu);
      }
    }
    __syncthreads();
  }

  // ======== score = sum_ij K_ij * exp(lu_i + lv_j) * sim_ij ========
  float partial = 0.0f;
#pragma unroll
  for (int t = 0; t < 32; ++t) {
    int idx = t * 128 + tid;
    int i = idx >> 6, j = idx & 63;
    partial += Ksh[i * STRIDE + j] * __expf(lu[i] + lv[j]) * simb[idx];
  }
  partial = wave_sum32(partial);
  if (lane == 0) red[wv] = partial;
  __syncthreads();
  if (tid == 0)
    out[b] = (red[0] + red[1] + red[2] + red[3]) * (TEMP / 64.0f);
}

extern "C" void kernel_launch(void* const* d_in, const int* in_sizes, int n_in,
                              void* d_out, int out_size, void* d_ws, size_t ws_size,
                              hipStream_t stream) {
  const float* sim = (const float*)d_in[0];
  const float* w1  = (const float*)d_in[1];
  const float* w2  = (const float*)d_in[2];
  float* out       = (float*)d_out;

  const int B = in_sizes[0] / (NDIM * NDIM);   // 1500
  emd_sinkhorn_kernel<<<B, 128, 0, stream>>>(sim, w1, w2, out);
}